// sparse3DBA_32349693673951
// MI455X (gfx1250) — compile-verified
//
#include <hip/hip_runtime.h>
#include <math.h>

typedef float v2f __attribute__((ext_vector_type(2)));
typedef float v8f __attribute__((ext_vector_type(8)));

// state layout in workspace (floats):
// [0..8]=R row-major, [9..11]=t, [12]=lam, [13]=prev_cost,
// [14..22]=R_candidate, [23..25]=t_candidate, [26]=nanflag
#define ST_R    0
#define ST_T    9
#define ST_LAM  12
#define ST_PC   13
#define ST_RN   14
#define ST_TN   23
#define ST_NAN  26

// ---------------------------------------------------------------------------
// Pass A: per-point Jacobian rows -> Gram accumulation via v_wmma_f32_16x16x4_f32
// Tile: 32 rows x 16 cols (stride 20 floats). Lane p (point base+p):
//   p<16  -> rows 2p,2p+1, cols 0..6  (col7 zero)
//   p>=16 -> rows 2(p-16),2(p-16)+1, cols 8..14 (col15 zero)
// 8 WMMAs consume the 32 rows; D accumulates X^T X with H_A in [0..6]^2 and
// H_B in [8..14]^2 (cross blocks ignored).
// ---------------------------------------------------------------------------
__global__ __launch_bounds__(256) void ba_accum(const float* __restrict__ pts,
                                                const float* __restrict__ fref,
                                                const float* __restrict__ Km,
                                                const float* __restrict__ state,
                                                float* __restrict__ partials,
                                                int N, int nTiles)
{
    __shared__ __align__(16) float tile[8][32 * 20];
    __shared__ float red[8][49];

    const int lane   = threadIdx.x & 31;
    const int wv     = threadIdx.x >> 5;
    const int waveId = blockIdx.x * 8 + wv;
    const int nWaves = gridDim.x * 8;

    const float fx = Km[0], cx = Km[2], fy = Km[4], cy = Km[5];
    float R[9], t[3];
#pragma unroll
    for (int i = 0; i < 9; ++i) R[i] = state[ST_R + i];
    t[0] = state[ST_T + 0]; t[1] = state[ST_T + 1]; t[2] = state[ST_T + 2];

    const int lp = lane & 15;          // column within half / point within half
    const int cb = (lane >> 4) * 8;    // column base: 0 or 8
    float* tw = tile[wv];

    v8f c;
#pragma unroll
    for (int i = 0; i < 8; ++i) c[i] = 0.0f;

    for (int tl = waveId; tl < nTiles; tl += nWaves) {
        const int idx = tl * 32 + lane;
        float r0v[8], r1v[8];
        if (idx < N) {
            const float x  = pts[idx * 3 + 0];
            const float y  = pts[idx * 3 + 1];
            const float z  = pts[idx * 3 + 2];
            const float f0 = fref[idx * 2 + 0];
            const float f1 = fref[idx * 2 + 1];
            // p = pts @ R + t  (p_j = sum_i pts_i R[i][j])
            const float p0 = x * R[0] + y * R[3] + z * R[6] + t[0];
            const float p1 = x * R[1] + y * R[4] + z * R[7] + t[1];
            const float p2 = x * R[2] + y * R[5] + z * R[8] + t[2];
            const float pr0 = fx * p0 + cx * p2;
            const float pr1 = fy * p1 + cy * p2;
            const int ui = (int)(pr0 / p2) - 1;   // trunc toward zero, like astype(int32)
            const int vi = (int)(pr1 / p2) - 1;
            const float e0 = (float)vi - f0;      // err = flip(pix) - feature_ref
            const float e1 = (float)ui - f1;
            const float iz = 1.0f / p2;
            const float a0 = fx * iz;
            const float a2 = -fx * p0 * iz * iz;
            const float b1 = fy * iz;
            const float b2 = -fy * p1 * iz * iz;
            // J row0 = [a0,0,a2 | a.(-skew(p)) cols]
            r0v[0] = a0;  r0v[1] = 0.0f; r0v[2] = a2;
            r0v[3] = a2 * p1;
            r0v[4] = a0 * p2 - a2 * p0;
            r0v[5] = -a0 * p1;
            r0v[6] = e0;  r0v[7] = 0.0f;
            // J row1 = [0,b1,b2 | b.(-skew(p)) cols]
            r1v[0] = 0.0f; r1v[1] = b1; r1v[2] = b2;
            r1v[3] = -b1 * p2 + b2 * p1;
            r1v[4] = -b2 * p0;
            r1v[5] = b1 * p0;
            r1v[6] = e1;  r1v[7] = 0.0f;
        } else {
#pragma unroll
            for (int i = 0; i < 8; ++i) { r0v[i] = 0.0f; r1v[i] = 0.0f; }
        }
        // stage rows into LDS (two float4 per row, 16B aligned: stride 20f=80B)
        float* q0 = &tw[(2 * lp + 0) * 20 + cb];
        float* q1 = &tw[(2 * lp + 1) * 20 + cb];
        ((float4*)q0)[0] = make_float4(r0v[0], r0v[1], r0v[2], r0v[3]);
        ((float4*)q0)[1] = make_float4(r0v[4], r0v[5], r0v[6], r0v[7]);
        ((float4*)q1)[0] = make_float4(r1v[0], r1v[1], r1v[2], r1v[3]);
        ((float4*)q1)[1] = make_float4(r1v[4], r1v[5], r1v[6], r1v[7]);
        __builtin_amdgcn_wave_barrier();   // order LDS stores vs loads (same wave, DS in-order)

        // 8 WMMAs: rows 4w..4w+3.  A lane layout (16x4 f32): v0 = K0(l<16)/K2(l>=16),
        // v1 = K1/K3; B (4x16) mirrors it, so one v2f serves both operands of the Gram.
        const int rbase = (lane >> 4) << 1;   // 0 or 2
#pragma unroll
        for (int w = 0; w < 8; ++w) {
            const int rr = 4 * w + rbase;
            v2f ab;
            ab.x = tw[(rr + 0) * 20 + lp];
            ab.y = tw[(rr + 1) * 20 + lp];
            c = __builtin_amdgcn_wmma_f32_16x16x4_f32(false, ab, false, ab,
                                                      (short)0, c, false, false);
        }
        __builtin_amdgcn_wave_barrier();
    }

    // Fold the two 7x7 diagonal Gram blocks: S[r][j] = C[r][j] + C[8+r][8+j].
    // C layout: vgpr r, lanes0-15 -> (M=r,N=lane); lanes16-31 -> (M=r+8,N=lane-16).
#pragma unroll
    for (int r = 0; r < 7; ++r) {
        const float v = c[r];
        const float o = __shfl(v, 24 + (lane & 7), 32);   // lane j<7 pulls C[8+r][8+j]
        const float s = v + o;
        if (lane < 7) red[wv][r * 7 + lane] = s;
    }
    __syncthreads();
    if (threadIdx.x < 49) {
        float s = 0.0f;
#pragma unroll
        for (int w = 0; w < 8; ++w) s += red[w][threadIdx.x];
        partials[blockIdx.x * 49 + threadIdx.x] = s;
    }
}

// ---------------------------------------------------------------------------
// Candidate cost: p = pts @ Rn.T + tn  (reference's transpose quirk replicated)
// ---------------------------------------------------------------------------
__global__ __launch_bounds__(256) void ba_cost(const float* __restrict__ pts,
                                               const float* __restrict__ fref,
                                               const float* __restrict__ Km,
                                               const float* __restrict__ state,
                                               float* __restrict__ partialC, int N)
{
    __shared__ float red[256];
    const float fx = Km[0], cx = Km[2], fy = Km[4], cy = Km[5];
    float Rn[9], tn[3];
#pragma unroll
    for (int i = 0; i < 9; ++i) Rn[i] = state[ST_RN + i];
    tn[0] = state[ST_TN + 0]; tn[1] = state[ST_TN + 1]; tn[2] = state[ST_TN + 2];

    float acc = 0.0f;
    for (int i = blockIdx.x * blockDim.x + threadIdx.x; i < N;
         i += gridDim.x * blockDim.x) {
        const float x = pts[i * 3 + 0], y = pts[i * 3 + 1], z = pts[i * 3 + 2];
        // p_j = sum_i pts_i * Rn[j][i]
        const float p0 = x * Rn[0] + y * Rn[1] + z * Rn[2] + tn[0];
        const float p1 = x * Rn[3] + y * Rn[4] + z * Rn[5] + tn[1];
        const float p2 = x * Rn[6] + y * Rn[7] + z * Rn[8] + tn[2];
        const int ui = (int)((fx * p0 + cx * p2) / p2) - 1;
        const int vi = (int)((fy * p1 + cy * p2) / p2) - 1;
        const float e0 = (float)vi - fref[i * 2 + 0];
        const float e1 = (float)ui - fref[i * 2 + 1];
        acc += 0.5f * (e0 * e0 + e1 * e1);
    }
    red[threadIdx.x] = acc;
    __syncthreads();
    for (int s = 128; s > 0; s >>= 1) {
        if (threadIdx.x < (unsigned)s) red[threadIdx.x] += red[threadIdx.x + s];
        __syncthreads();
    }
    if (threadIdx.x == 0) partialC[blockIdx.x] = red[0];
}

// ---------------------------------------------------------------------------
// Reduce Gram partials, build LM system, solve 6x6, SO3 exp, pose candidate.
// ---------------------------------------------------------------------------
__global__ void ba_solve(float* __restrict__ state,
                         const float* __restrict__ partials, int nblk)
{
    __shared__ float S[49];
    const int tid = threadIdx.x;
    if (tid < 49) {
        float s = 0.0f;
        for (int b = 0; b < nblk; ++b) s += partials[b * 49 + tid];
        S[tid] = s;
    }
    __syncthreads();
    if (tid != 0) return;

    float H[6][6], g[6];
    for (int i = 0; i < 6; ++i) {
        for (int j = 0; j < 6; ++j) H[i][j] = S[i * 7 + j];
        g[i] = S[i * 7 + 6];
    }
    const float lam = state[ST_LAM];
    for (int i = 0; i < 6; ++i)
        H[i][i] = H[i][i] + (H[i][i] + 1e-9f) * lam;

    // Gauss-Jordan inverse with partial pivoting: M = [Hlm | I]
    float M[6][12];
    for (int i = 0; i < 6; ++i)
        for (int k = 0; k < 12; ++k)
            M[i][k] = (k < 6) ? H[i][k] : ((k - 6 == i) ? 1.0f : 0.0f);
    for (int col = 0; col < 6; ++col) {
        int piv = col;
        float mx = fabsf(M[col][col]);
        for (int r = col + 1; r < 6; ++r) {
            const float a = fabsf(M[r][col]);
            if (a > mx) { mx = a; piv = r; }
        }
        if (piv != col)
            for (int k = 0; k < 12; ++k) {
                const float tmp = M[col][k]; M[col][k] = M[piv][k]; M[piv][k] = tmp;
            }
        const float inv = 1.0f / M[col][col];
        for (int k = 0; k < 12; ++k) M[col][k] *= inv;
        for (int r = 0; r < 6; ++r) {
            if (r == col) continue;
            const float f = M[r][col];
            for (int k = 0; k < 12; ++k) M[r][k] -= f * M[col][k];
        }
    }
    float delta[6];
    int nanflag = 0;
    for (int k = 0; k < 6; ++k) {
        float s = 0.0f;
        for (int j = 0; j < 6; ++j) s += M[k][6 + j] * g[j];
        delta[k] = -s;
        if (delta[k] != delta[k]) nanflag = 1;
    }

    // dr = so3exp(dw)
    const float w0 = delta[3], w1 = delta[4], w2 = delta[5];
    const float th = sqrtf(w0 * w0 + w1 * w1 + w2 * w2);
    const bool small = th < 1e-7f;
    const float a = small ? 1.0f : (sinf(th) / th);
    const float b = small ? 0.5f : ((1.0f - cosf(th)) / (th * th));
    const float W[9]  = { 0.0f, -w2, w1,  w2, 0.0f, -w0,  -w1, w0, 0.0f };
    float W2[9];
    for (int i = 0; i < 3; ++i)
        for (int j = 0; j < 3; ++j) {
            float s = 0.0f;
            for (int k = 0; k < 3; ++k) s += W[i * 3 + k] * W[k * 3 + j];
            W2[i * 3 + j] = s;
        }
    float dr[9];
    for (int i = 0; i < 3; ++i)
        for (int j = 0; j < 3; ++j)
            dr[i * 3 + j] = ((i == j) ? 1.0f : 0.0f) + a * W[i * 3 + j] + b * W2[i * 3 + j];

    // Rn = dr @ R ; tn = dr @ t + dt
    for (int i = 0; i < 3; ++i)
        for (int j = 0; j < 3; ++j) {
            float s = 0.0f;
            for (int k = 0; k < 3; ++k) s += dr[i * 3 + k] * state[ST_R + k * 3 + j];
            state[ST_RN + i * 3 + j] = s;
        }
    for (int i = 0; i < 3; ++i) {
        float s = delta[i];
        for (int k = 0; k < 3; ++k) s += dr[i * 3 + k] * state[ST_T + k];
        state[ST_TN + i] = s;
    }
    state[ST_NAN] = (float)nanflag;
}

// ---------------------------------------------------------------------------
// Reduce cost partials, accept/reject, update lambda.
// ---------------------------------------------------------------------------
__global__ void ba_decide(float* __restrict__ state,
                          const float* __restrict__ partialC,
                          int nblk, int N, int is_init)
{
    __shared__ float red[256];
    float s = 0.0f;
    for (int i = threadIdx.x; i < nblk; i += 256) s += partialC[i];
    red[threadIdx.x] = s;
    __syncthreads();
    for (int st = 128; st > 0; st >>= 1) {
        if (threadIdx.x < (unsigned)st) red[threadIdx.x] += red[threadIdx.x + st];
        __syncthreads();
    }
    if (threadIdx.x != 0) return;

    const float newc = red[0] / (float)N;
    if (is_init) {
        state[ST_PC] = newc;
        return;
    }
    const float prev  = state[ST_PC];
    const bool worse  = newc > prev;                 // NaN > x == false, like JAX
    const bool bad    = worse || (state[ST_NAN] != 0.0f);
    float lam = state[ST_LAM] * (worse ? 10.0f : 0.1f);
    lam = fminf(fmaxf(lam, 1e-6f), 1e4f);
    state[ST_LAM] = lam;
    if (!bad) {
        for (int i = 0; i < 9; ++i) state[ST_R + i] = state[ST_RN + i];
        for (int i = 0; i < 3; ++i) state[ST_T + i] = state[ST_TN + i];
        state[ST_PC] = newc;
    }
}

__global__ void ba_init(float* state)
{
    if (threadIdx.x == 0) {
        for (int i = 0; i < 3; ++i)
            for (int j = 0; j < 3; ++j) {
                const float v = (i == j) ? 1.0f : 0.0f;
                state[ST_R + i * 3 + j]  = v;
                state[ST_RN + i * 3 + j] = v;
            }
        state[ST_T + 0] = 1.0f; state[ST_T + 1] = 1.0f; state[ST_T + 2] = 0.0f;
        state[ST_TN + 0] = 1.0f; state[ST_TN + 1] = 1.0f; state[ST_TN + 2] = 0.0f;
        state[ST_LAM] = 0.01f;
        state[ST_PC]  = 0.0f;
        state[ST_NAN] = 0.0f;
    }
}

__global__ void ba_emit(const float* __restrict__ state, float* __restrict__ out)
{
    const int i = threadIdx.x;
    if (i < 12) out[i] = state[i];   // R row-major [0..8], then t [9..11]
}

// ---------------------------------------------------------------------------
extern "C" void kernel_launch(void* const* d_in, const int* in_sizes, int n_in,
                              void* d_out, int out_size, void* d_ws, size_t ws_size,
                              hipStream_t stream)
{
    const float* pts  = (const float*)d_in[0];   // (N,3)
    const float* fref = (const float*)d_in[1];   // (N,2)
    const float* Km   = (const float*)d_in[5];   // (3,3)
    const int N = in_sizes[0] / 3;

    float* state = (float*)d_ws;
    size_t wsf = ws_size / sizeof(float);
    int nA = 512, nC = 512;
    if (wsf < (size_t)(32 + 49 * nA + nC)) { nA = 64; nC = 64; }
    if (wsf < (size_t)(32 + 49 * nA + nC)) { nA = 8;  nC = 8;  }
    float* pA = state + 32;
    float* pC = pA + (size_t)nA * 49;

    const int nTiles = (N + 31) / 32;

    ba_init<<<1, 32, 0, stream>>>(state);
    ba_cost<<<nC, 256, 0, stream>>>(pts, fref, Km, state, pC, N);
    ba_decide<<<1, 256, 0, stream>>>(state, pC, nC, N, 1);

    for (int it = 0; it < 10; ++it) {
        ba_accum<<<nA, 256, 0, stream>>>(pts, fref, Km, state, pA, N, nTiles);
        ba_solve<<<1, 64, 0, stream>>>(state, pA, nA);
        ba_cost<<<nC, 256, 0, stream>>>(pts, fref, Km, state, pC, N);
        ba_decide<<<1, 256, 0, stream>>>(state, pC, nC, N, 0);
    }
    ba_emit<<<1, 32, 0, stream>>>(state, (float*)d_out);
}